// CombinedLoss_678604832860
// MI455X (gfx1250) — compile-verified
//
#include <hip/hip_runtime.h>
#include <math.h>

#define BATCH   32
#define NPTS    200000
#define TPB     256
#define TILE    1024                       // TPB * 4 points per tile
#define NBLK    64                         // blocks per batch
#define NTILES  ((NPTS + TILE - 1) / TILE) // 196

// workspace layout (in floats)
#define MG_OFF   0     // 32 * 12  : ground-truth 3x4 combined matrices
#define MP_OFF   384   // 32 * 12  : predicted    3x4 combined matrices
#define POSE_OFF 768   // 32       : per-batch pose loss
#define PART_OFF 800   // float2[32*64] partial (sumW, sumErrW)

typedef float v2f __attribute__((ext_vector_type(2)));
typedef float v8f __attribute__((ext_vector_type(8)));

// ---------------------------------------------------------------------------
// CDNA5 async global->LDS (16B per lane), ASYNCcnt-tracked.
// ---------------------------------------------------------------------------
__device__ __forceinline__ void async_ld16(const float* g, float* l) {
  unsigned lo = (unsigned)(size_t)l;   // low 32 bits of generic LDS addr = LDS byte offset
  asm volatile("global_load_async_to_lds_b128 %0, %1, off"
               :: "v"(lo), "v"(g) : "memory");
}

// ---------------------------------------------------------------------------
// Kernel 1: per-batch pose loss (smooth-L1 + quaternion distance). 1 wave.
// ---------------------------------------------------------------------------
__global__ void prep_pose(const float* __restrict__ tt, const float* __restrict__ tr,
                          const float* __restrict__ te, const float* __restrict__ re,
                          float* __restrict__ ws) {
  int b = threadIdx.x;
  if (b >= BATCH) return;
  float sl1 = 0.f;
  #pragma unroll
  for (int i = 0; i < 3; ++i) {
    float d  = te[b * 3 + i] - tt[b * 3 + i];
    float ad = fabsf(d);
    sl1 += (ad < 1.f) ? 0.5f * d * d : (ad - 0.5f);
  }
  float q0 = re[b*4+0], q1 = re[b*4+1], q2 = re[b*4+2], q3 = re[b*4+3];
  float inq = __builtin_amdgcn_rsqf(q0*q0 + q1*q1 + q2*q2 + q3*q3);
  q0 *= inq; q1 *= inq; q2 *= inq; q3 *= inq;
  float r0 = tr[b*4+0], r1 = tr[b*4+1], r2 = tr[b*4+2], r3 = tr[b*4+3];
  float inr = __builtin_amdgcn_rsqf(r0*r0 + r1*r1 + r2*r2 + r3*r3);
  r0 *= inr; r1 = -r1 * inr; r2 = -r2 * inr; r3 = -r3 * inr;   // conjugate
  float w = q0*r0 - q1*r1 - q2*r2 - q3*r3;
  float x = q0*r1 + q1*r0 + q2*r3 - q3*r2;
  float y = q0*r2 - q1*r3 + q2*r0 + q3*r1;
  float z = q0*r3 + q1*r2 - q2*r1 + q3*r0;
  float dist = 2.f * atan2f(sqrtf(x*x + y*y + z*z), fabsf(w));
  ws[POSE_OFF + b] = dist + sl1;   // RESCALE_ROT = RESCALE_TRANS = 1
}

// ---------------------------------------------------------------------------
// Kernel 2: compose M = X1 @ RT (3x4 kept) for all 64 (batch,pose) pairs.
// WMMA path: one V_WMMA_F32_16X16X4_F32 computes 4 products at once as the
// block-diagonal of the 16x16 result (A rows = 4 stacked X1s, B cols = 4 RTs).
// ---------------------------------------------------------------------------
__device__ __forceinline__ float x1_elem(const float* cam, int b, int r, int k) {
  if (r == 3) return (k == 3) ? 1.f : 0.f;
  if (k == 3) return 0.f;
  return cam[b * 9 + r * 3 + k];
}

__device__ __forceinline__ float rt_elem(int i, int j, float w, float x, float y,
                                         float z, float tx, float ty, float tz) {
  if (i == 3) return (j == 3) ? 1.f : 0.f;
  if (j == 3) return (i == 0) ? tx : ((i == 1) ? ty : tz);
  float R[9] = {
    1.f - 2.f*(y*y + z*z), 2.f*(x*y - z*w),       2.f*(x*z + y*w),
    2.f*(x*y + z*w),       1.f - 2.f*(x*x + z*z), 2.f*(y*z - x*w),
    2.f*(x*z - y*w),       2.f*(y*z + x*w),       1.f - 2.f*(x*x + y*y)};
  return R[i * 3 + j];
}

__global__ void compose_mats(const float* __restrict__ cam, const float* __restrict__ tt,
                             const float* __restrict__ tr, const float* __restrict__ te,
                             const float* __restrict__ re, float* __restrict__ ws) {
#if __has_builtin(__builtin_amdgcn_wmma_f32_16x16x4_f32)
  int lane = threadIdx.x;          // launched with exactly 32 threads, EXEC all 1s
  int m  = lane & 15;
  int hi = lane >> 4;
  int k0 = hi ? 2 : 0;             // K split 0,1 / 2,3 across lane halves (A & B)
  int q4 = m >> 2;                 // which of the 4 products in this group
  int j  = m & 3;                  // row of X1 (for A) == col of RT (for B)
  for (int g = 0; g < 16; ++g) {   // 16 WMMAs cover 64 products
    int p  = 4 * g + q4;
    int bb = p >> 1;
    int wh = p & 1;                // 0 = ground truth, 1 = predicted
    v2f a;
    a.x = x1_elem(cam, bb, j, k0);
    a.y = x1_elem(cam, bb, j, k0 + 1);
    const float* qs = wh ? re : tr;
    const float* ts = wh ? te : tt;
    float qw = qs[bb*4+0], qx = qs[bb*4+1], qy = qs[bb*4+2], qz = qs[bb*4+3];
    float inr = __builtin_amdgcn_rsqf(qw*qw + qx*qx + qy*qy + qz*qz);
    qw *= inr; qx *= inr; qy *= inr; qz *= inr;
    float tx = ts[bb*3+0], ty = ts[bb*3+1], tz = ts[bb*3+2];
    v2f bv;
    bv.x = rt_elem(k0,     j, qw, qx, qy, qz, tx, ty, tz);
    bv.y = rt_elem(k0 + 1, j, qw, qx, qy, qz, tx, ty, tz);
    v8f c = {};
    v8f d = __builtin_amdgcn_wmma_f32_16x16x4_f32(false, a, false, bv,
                                                  (short)0, c, false, false);
    float dd[8];
    #pragma unroll
    for (int i = 0; i < 8; ++i) dd[i] = d[i];
    // D layout: lanes 0-15 hold rows 0-7, lanes 16-31 hold rows 8-15 (N = lane%16)
    bool valid = hi ? (q4 >= 2) : (q4 < 2);
    int  basev = hi ? (4 * q4 - 8) : (4 * q4);
    if (valid) {
      float* M = ws + (wh ? MP_OFF : MG_OFF) + bb * 12;
      M[0 * 4 + j] = dd[basev + 0];
      M[1 * 4 + j] = dd[basev + 1];
      M[2 * 4 + j] = dd[basev + 2];
    }
  }
#else
  int b = threadIdx.x;
  if (b >= BATCH) return;
  for (int wh = 0; wh < 2; ++wh) {
    const float* qs = wh ? re : tr;
    const float* ts = wh ? te : tt;
    float qw = qs[b*4+0], qx = qs[b*4+1], qy = qs[b*4+2], qz = qs[b*4+3];
    float inr = __builtin_amdgcn_rsqf(qw*qw + qx*qx + qy*qy + qz*qz);
    qw *= inr; qx *= inr; qy *= inr; qz *= inr;
    float tx = ts[b*3+0], ty = ts[b*3+1], tz = ts[b*3+2];
    float* M = ws + (wh ? MP_OFF : MG_OFF) + b * 12;
    for (int i = 0; i < 3; ++i)
      for (int jj = 0; jj < 4; ++jj) {
        float acc = 0.f;
        for (int k = 0; k < 4; ++k)
          acc += x1_elem(cam, b, i, k) * rt_elem(k, jj, qw, qx, qy, qz, tx, ty, tz);
        M[i * 4 + jj] = acc;
      }
  }
#endif
}

// ---------------------------------------------------------------------------
// Kernel 3: streaming point-cloud loss. Async-LDS double-buffered, wave-
// synchronous (each lane consumes only the LDS bytes its own async load wrote).
// ---------------------------------------------------------------------------
__global__ __launch_bounds__(TPB) void pc_loss_kernel(const float* __restrict__ pc,
                                                      const float* __restrict__ cam,
                                                      const float* __restrict__ ws,
                                                      float2* __restrict__ part) {
  __shared__ float sx[2][TILE], sy[2][TILE], sz[2][TILE];
  __shared__ float rW[TPB], rE[TPB];
  const int b = blockIdx.y, blk = blockIdx.x, tid = threadIdx.x;

  float mg[12], mp[12];
  #pragma unroll
  for (int i = 0; i < 12; ++i) mg[i] = ws[MG_OFF + b * 12 + i];
  #pragma unroll
  for (int i = 0; i < 12; ++i) mp[i] = ws[MP_OFF + b * 12 + i];
  const float cx = cam[b * 9 + 2], cy = cam[b * 9 + 5];

  const float* xr = pc + ((size_t)b * 4 + 0) * NPTS;
  const float* yr = xr + NPTS;
  const float* zr = yr + NPTS;

  float accW = 0.f, accE = 0.f;
  int buf = 0;
  const int t0 = blk;
  if (t0 < NTILES) {
    int base = t0 * TILE + tid * 4;
    async_ld16(xr + base, &sx[buf][tid * 4]);
    async_ld16(yr + base, &sy[buf][tid * 4]);
    async_ld16(zr + base, &sz[buf][tid * 4]);
  }
  for (int t = t0; t < NTILES; t += NBLK) {
    const int nt = t + NBLK;
    if (nt < NTILES) {
      int nb = nt * TILE + tid * 4;
      async_ld16(xr + nb, &sx[buf ^ 1][tid * 4]);
      async_ld16(yr + nb, &sy[buf ^ 1][tid * 4]);
      async_ld16(zr + nb, &sz[buf ^ 1][tid * 4]);
      asm volatile("s_wait_asynccnt 0x3" ::: "memory");  // current tile's 3 done
    } else {
      asm volatile("s_wait_asynccnt 0x0" ::: "memory");
    }
    float4 x4 = *(const float4*)&sx[buf][tid * 4];
    float4 y4 = *(const float4*)&sy[buf][tid * 4];
    float4 z4 = *(const float4*)&sz[buf][tid * 4];
    float xs[4] = {x4.x, x4.y, x4.z, x4.w};
    float ys[4] = {y4.x, y4.y, y4.z, y4.w};
    float zs[4] = {z4.x, z4.y, z4.z, z4.w};
    const int base = t * TILE + tid * 4;
    #pragma unroll
    for (int i = 0; i < 4; ++i) {
      float X = xs[i], Y = ys[i], Z = zs[i];
      float g0 = fmaf(mg[0], X, fmaf(mg[1], Y, fmaf(mg[2],  Z, mg[3])));
      float g1 = fmaf(mg[4], X, fmaf(mg[5], Y, fmaf(mg[6],  Z, mg[7])));
      float g2 = fmaf(mg[8], X, fmaf(mg[9], Y, fmaf(mg[10], Z, mg[11])));
      float p0 = fmaf(mp[0], X, fmaf(mp[1], Y, fmaf(mp[2],  Z, mp[3])));
      float p1 = fmaf(mp[4], X, fmaf(mp[5], Y, fmaf(mp[6],  Z, mp[7])));
      float p2 = fmaf(mp[8], X, fmaf(mp[9], Y, fmaf(mp[10], Z, mp[11])));
      float ig = __builtin_amdgcn_rcpf(g2);
      float ip = __builtin_amdgcn_rcpf(p2);
      float Fg = g0 * ig, Sg = g1 * ig;
      float F1 = p0 * ip, S1 = p1 * ip;
      float dx = Fg - cx, dy = Sg - cy;
      float w = __builtin_amdgcn_rsqf(fmaf(dx, dx, dy * dy));
      float F   = (Fg  < 1280.f) ? Fg  : F1;  F   = (F   > 0.f) ? F   : F1;
      float F1m = (F1  > 0.f)    ? F1  : F;   F1m = (F1m < 1280.f) ? F1m : F;
      float S   = (Sg  < 384.f)  ? Sg  : S1;  S   = (S   > 0.f) ? S   : S1;
      float S1m = (S1  > 0.f)    ? S1  : S;   S1m = (S1m < 384.f) ? S1m : S;
      float eF = F - F1m, eS = S - S1m;
      float err = __builtin_amdgcn_sqrtf(fmaf(eF, eF, eS * eS)) * w;
      if (base + i < NPTS) { accW += w; accE += err; }   // mask tail, selects only
    }
    buf ^= 1;
  }
  rW[tid] = accW; rE[tid] = accE;
  __syncthreads();
  for (int s = TPB >> 1; s > 0; s >>= 1) {
    if (tid < s) { rW[tid] += rW[tid + s]; rE[tid] += rE[tid + s]; }
    __syncthreads();
  }
  if (tid == 0) part[b * NBLK + blk] = make_float2(rW[0], rE[0]);
}

// ---------------------------------------------------------------------------
// Kernel 4: deterministic finalize.
// pc_b = sum(err*w) / (N * max(sum(w), 5));  total = 0.5*pose + 0.5*pc
// ---------------------------------------------------------------------------
__global__ void finalize(const float* __restrict__ ws, const float2* __restrict__ part,
                         float* __restrict__ out) {
  __shared__ float sp[BATCH], sq[BATCH];
  int tid = threadIdx.x;
  if (tid < BATCH) {
    float A = 0.f, C = 0.f;
    for (int k = 0; k < NBLK; ++k) {
      float2 p = part[tid * NBLK + k];
      A += p.x; C += p.y;
    }
    sp[tid] = C / ((float)NPTS * fmaxf(A, 5.f));
    sq[tid] = ws[POSE_OFF + tid];
  }
  __syncthreads();
  if (tid == 0) {
    float a = 0.f, p = 0.f;
    for (int i = 0; i < BATCH; ++i) { a += sp[i]; p += sq[i]; }
    out[0] = 0.5f * (p / (float)BATCH) + 0.5f * (a / (float)BATCH);
  }
}

extern "C" void kernel_launch(void* const* d_in, const int* in_sizes, int n_in,
                              void* d_out, int out_size, void* d_ws, size_t ws_size,
                              hipStream_t stream) {
  const float* pc  = (const float*)d_in[0];
  const float* tt  = (const float*)d_in[1];
  const float* tr  = (const float*)d_in[2];
  const float* te  = (const float*)d_in[3];
  const float* re  = (const float*)d_in[4];
  const float* cam = (const float*)d_in[5];
  float*  ws   = (float*)d_ws;
  float2* part = (float2*)(ws + PART_OFF);

  prep_pose   <<<1, 32, 0, stream>>>(tt, tr, te, re, ws);
  compose_mats<<<1, 32, 0, stream>>>(cam, tt, tr, te, re, ws);
  pc_loss_kernel<<<dim3(NBLK, BATCH), TPB, 0, stream>>>(pc, cam, ws, part);
  finalize    <<<1, 32, 0, stream>>>(ws, part, (float*)d_out);
}